// RelDecoder_39127152066939
// MI455X (gfx1250) — compile-verified
//
#include <hip/hip_runtime.h>

typedef __attribute__((ext_vector_type(2))) float v2f;
typedef __attribute__((ext_vector_type(4))) float v4f;
typedef __attribute__((ext_vector_type(8))) float v8f;

#define D_DIM           128
#define REL_ELEMS       (64 * 128)     // 32 KB fp32 relation table -> LDS
#define EDGES_PER_WAVE  16
#define WAVES_PER_BLOCK 8
#define BLOCK_THREADS   (WAVES_PER_BLOCK * 32)

// DistMult edge scorer:
//   out[e] = sigmoid( sum_d z[src[e],d] * rel[rel_id[e],d] * z[dst[e],d] )
//
// Wave32: 16 edges per wave. Lanes n and n+16 split edge n's 128 dims and
// accumulate fp32 partials into the two VGPRs that form column n of the
// 4x16 fp32 B-matrix. One V_WMMA_F32_16X16X4_F32 with A = ones performs the
// cross-lane reduction: D[i][n] = column-sum of B = score(edge n), replicated
// down every row — so acc[0] holds all 16 scores striped across lanes, and
// the epilogue is a fully parallel sigmoid + one coalesced 64B store.
__global__ __launch_bounds__(BLOCK_THREADS) void distmult_wmma_kernel(
    const float* __restrict__ z,
    const long long* __restrict__ edge_src,
    const long long* __restrict__ edge_dst,
    const long long* __restrict__ rel_id,
    const float* __restrict__ rel,
    float* __restrict__ out,
    int n_edges)
{
    __shared__ __align__(16) float lds_rel[REL_ELEMS];

    // Cooperative b128 preload of the 32 KB relation table into LDS.
    for (int i = threadIdx.x * 4; i < REL_ELEMS; i += BLOCK_THREADS * 4)
        *(v4f*)(lds_rel + i) = *(const v4f*)(rel + i);
    __syncthreads();

    const int lane = threadIdx.x & 31;
    const int n    = lane & 15;        // edge slot (B-matrix column)
    const int half = lane >> 4;        // 0: dims 0..63, 1: dims 64..127

    const long long wave_id   = (long long)blockIdx.x * WAVES_PER_BLOCK
                              + (threadIdx.x >> 5);
    const long long edge_base = wave_id * EDGES_PER_WAVE;
    if (edge_base >= n_edges) return;  // wave-uniform: EXEC stays all-ones

    long long e = edge_base + n;
    if (e >= n_edges) e = n_edges - 1; // branchless tail clamp (EXEC full)

    const float* sp = z       + (long long)edge_src[e] * D_DIM + half * 64;
    const float* dp = z       + (long long)edge_dst[e] * D_DIM + half * 64;
    const float* rp = lds_rel + (int)rel_id[e]         * D_DIM + half * 64;

    // Per-lane partials: these two floats ARE this lane's B-matrix slots
    // for column n (which K indices they map to is irrelevant — all summed).
    v2f b = {0.0f, 0.0f};
#pragma unroll 4
    for (int i = 0; i < 16; ++i) {
        v4f s = *(const v4f*)(sp + 4 * i);   // global_load_b128 (L2-resident z)
        v4f d = *(const v4f*)(dp + 4 * i);   // global_load_b128
        v4f r = *(const v4f*)(rp + 4 * i);   // ds_load_b128
        b.x = fmaf(s.x * r.x, d.x, b.x);
        b.y = fmaf(s.y * r.y, d.y, b.y);
        b.x = fmaf(s.z * r.z, d.z, b.x);
        b.y = fmaf(s.w * r.w, d.w, b.y);
    }

    // A = ones (16x4): D[i][n] = sum_k B[k][n] -> full fp32 dot per edge,
    // broadcast across all rows i.
    v2f aones = {1.0f, 1.0f};
    v8f acc   = {0.f, 0.f, 0.f, 0.f, 0.f, 0.f, 0.f, 0.f};
    acc = __builtin_amdgcn_wmma_f32_16x16x4_f32(
        /*neg_a=*/false, aones, /*neg_b=*/false, b,
        /*c_mod=*/(short)0, acc, /*reuse_a=*/false, /*reuse_b=*/false);

    // Lane l holds score(edge_base + (l&15)) in acc[0]. Parallel sigmoid via
    // v_exp_f32 + v_rcp_f32 (no IEEE-divide expansion), coalesced store from
    // the low half-wave.
    const float score = acc[0];
    const float sig   = __builtin_amdgcn_rcpf(1.0f + __expf(-score));

    const long long eo = edge_base + lane;
    if (lane < 16 && eo < n_edges)
        out[eo] = sig;
}

extern "C" void kernel_launch(void* const* d_in, const int* in_sizes, int n_in,
                              void* d_out, int out_size, void* d_ws, size_t ws_size,
                              hipStream_t stream) {
    const float*     z          = (const float*)d_in[0];
    const long long* edge_index = (const long long*)d_in[1]; // [2, E] int64
    const long long* rel_id     = (const long long*)d_in[2]; // [E]    int64
    const float*     rel        = (const float*)d_in[3];     // [64, 128]
    float*           out        = (float*)d_out;

    const int n_edges = in_sizes[2];
    const long long* edge_src = edge_index;
    const long long* edge_dst = edge_index + n_edges;

    const int edges_per_block = WAVES_PER_BLOCK * EDGES_PER_WAVE; // 128
    const int grid = (n_edges + edges_per_block - 1) / edges_per_block;

    distmult_wmma_kernel<<<grid, BLOCK_THREADS, 0, stream>>>(
        z, edge_src, edge_dst, rel_id, rel, out, n_edges);
}